// CombinedModel_72464688218553
// MI455X (gfx1250) — compile-verified
//
#include <hip/hip_runtime.h>
#include <hip/hip_bf16.h>
#include <cstdint>
#include <cstddef>

typedef __attribute__((ext_vector_type(2))) float v2f;
typedef __attribute__((ext_vector_type(8))) float v8f;

#define D_FEAT 128   // feature width of x / h / hid (all 128 in this model)

// ---------------------------------------------------------------------------
// deg[dst] += 1 per edge
// ---------------------------------------------------------------------------
__global__ void degree_kernel(const int* __restrict__ ei, float* __restrict__ deg, int E) {
    int i = blockIdx.x * blockDim.x + threadIdx.x;
    int stride = gridDim.x * blockDim.x;
    for (int e = i; e < E; e += stride) {
        atomicAdd(&deg[ei[(size_t)E + e]], 1.0f);
    }
}

// ---------------------------------------------------------------------------
// out[dst] += feat[src]  (128 channels). One wave per edge, float4 per lane.
// ---------------------------------------------------------------------------
__global__ void scatter_kernel(const float* __restrict__ feat, const int* __restrict__ ei,
                               float* __restrict__ out, int E) {
    int tid  = blockIdx.x * blockDim.x + threadIdx.x;
    int gw   = tid >> 5;
    int lane = tid & 31;
    int nw   = (gridDim.x * blockDim.x) >> 5;
    for (int e = gw; e < E; e += nw) {
        int s = ei[e];
        int d = ei[(size_t)E + e];
        float4 v = ((const float4*)(feat + (size_t)s * D_FEAT))[lane];
        float* op = out + (size_t)d * D_FEAT + lane * 4;
        atomicAdd(op + 0, v.x);
        atomicAdd(op + 1, v.y);
        atomicAdd(op + 2, v.z);
        atomicAdd(op + 3, v.w);
    }
}

// ---------------------------------------------------------------------------
// agg[row,:] /= max(deg[row],1)   (row width 128, float4 granularity)
// ---------------------------------------------------------------------------
__global__ void normalize_kernel(float* __restrict__ agg, const float* __restrict__ deg, int M) {
    int i = blockIdx.x * blockDim.x + threadIdx.x;
    int total  = M * (D_FEAT / 4);   // float4 elements
    int stride = gridDim.x * blockDim.x;
    for (int j = i; j < total; j += stride) {
        int row = j >> 5;            // j / (128/4)
        float inv = 1.0f / fmaxf(deg[row], 1.0f);
        float4* p = (float4*)agg + j;
        float4 v = *p;
        v.x *= inv; v.y *= inv; v.z *= inv; v.w *= inv;
        *p = v;
    }
}

// ---------------------------------------------------------------------------
// C[M,N] = act(A[M,K] @ W[K,N] + bias)  via v_wmma_f32_16x16x4_f32.
// N = NTN*16 (compile-time). W is staged TRANSPOSED into LDS once per block
// (ldsw[n*K+k]) so each B fragment is one contiguous ds_load_b64. Each wave
// owns one 16-row M tile and keeps all NTN 16x16 accumulators in registers,
// so one A-fragment global load feeds NTN WMMAs.
// GATHER: A row m is taken from A[(*basep) + idx[m]] (row gather).
// Fragment layouts per CDNA5 ISA 7.12.2:
//   A: lane -> (M = lane&15, K = (lane>>4)*2 + j)      j = 0,1
//   B: lane -> (N = lane&15, K = (lane>>4)*2 + j)      j = 0,1
//   D: VGPR j -> row (lane>>4)*8 + j, col lane&15
// ---------------------------------------------------------------------------
template <int NTN, bool RELU, bool GATHER>
__global__ __launch_bounds__(256) void
gemm_wmma_kernel(const float* __restrict__ A, const float* __restrict__ W,
                 const float* __restrict__ bias, float* __restrict__ C,
                 int M, int K,
                 const int* __restrict__ idx, const int* __restrict__ basep) {
    constexpr int N = NTN * 16;
    __shared__ float ldsw[N * 128];          // K <= 128

    const int tid  = threadIdx.x;
    const int lane = tid & 31;

    // Stage W transposed: ldsw[n*K + k] = W[k*N + n]  (coalesced global reads)
    for (int i = tid; i < K * N; i += 256) {
        int k = i / N;
        int n = i - k * N;
        ldsw[n * K + k] = W[i];
    }
    __syncthreads();

    const int mtiles = M >> 4;
    const int wave   = blockIdx.x * 8 + (tid >> 5);
    if (wave >= mtiles) return;              // uniform per wave: EXEC stays all-1s

    const int half = lane >> 4;              // 0 or 1
    const int l16  = lane & 15;

    const int ma = wave * 16 + l16;
    size_t rowA;
    if (GATHER) {
        rowA = (size_t)(basep[0] + idx[ma]);
    } else {
        rowA = (size_t)ma;
    }
    const float* arow = A + rowA * (size_t)K + half * 2;

    v8f acc[NTN];
#pragma unroll
    for (int tn = 0; tn < NTN; ++tn) acc[tn] = {};

    for (int k = 0; k < K; k += 4) {
        v2f a;
        a.x = arow[k];
        a.y = arow[k + 1];
#pragma unroll
        for (int tn = 0; tn < NTN; ++tn) {
            int n = tn * 16 + l16;
            const float* bp = ldsw + n * K + k + half * 2;
            v2f b;
            b.x = bp[0];
            b.y = bp[1];                     // contiguous -> ds_load_b64
            acc[tn] = __builtin_amdgcn_wmma_f32_16x16x4_f32(
                false, a, false, b, (short)0, acc[tn], false, false);
        }
    }

    const int rbase = wave * 16 + half * 8;
#pragma unroll
    for (int tn = 0; tn < NTN; ++tn) {
        int n = tn * 16 + l16;
        float bv = bias ? bias[n] : 0.0f;
#pragma unroll
        for (int j = 0; j < 8; ++j) {
            float v = acc[tn][j] + bv;
            if (RELU) v = fmaxf(v, 0.0f);
            C[(size_t)(rbase + j) * N + n] = v;
        }
    }
}

// ---------------------------------------------------------------------------
// For each pair p = t*G + g:
//   hid = relu(Atf[t,:] + Bg[g,:])        (128, bias bm1 prefolded into Atf)
//   o_c = hid . Wm2[:,c] + bm2[c]         (c = 0,1);  out[p,:] = softmax(o)
// Block = 8 waves, all with the same t (Atf/Wm2/bm2 hoisted out of the loop);
// each wave streams a g-chunk, one float4 of Bg per lane per pair.
// Grid = T * GCHUNKS blocks.
// ---------------------------------------------------------------------------
#define GCHUNKS 8
__global__ __launch_bounds__(256) void
pair_kernel(const float* __restrict__ Atf, const float* __restrict__ Bg,
            const float* __restrict__ Wm2, const float* __restrict__ bm2,
            float* __restrict__ out, int T, int G) {
    const int lane = threadIdx.x & 31;
    const int wv   = threadIdx.x >> 5;
    const int t    = blockIdx.x / GCHUNKS;
    const int gc   = blockIdx.x - t * GCHUNKS;
    const int span = (G + GCHUNKS - 1) / GCHUNKS;
    const int g0   = gc * span;
    const int g1   = (g0 + span < G) ? (g0 + span) : G;

    // loop-invariant per-lane data
    const float4 a   = ((const float4*)(Atf + (size_t)t * D_FEAT))[lane];
    const float4 w01 = ((const float4*)Wm2)[lane * 2 + 0]; // (W[k,0],W[k,1],W[k+1,0],W[k+1,1])
    const float4 w23 = ((const float4*)Wm2)[lane * 2 + 1];
    const float  c0  = bm2[0];
    const float  c1  = bm2[1];

    for (int g = g0 + wv; g < g1; g += 8) {
        const float4* brow = (const float4*)(Bg + (size_t)g * D_FEAT);
        __builtin_prefetch(Bg + (size_t)(g + 8) * D_FEAT, 0, 0); // global_prefetch_b8
        float4 b = brow[lane];
        float h0 = fmaxf(a.x + b.x, 0.0f);
        float h1 = fmaxf(a.y + b.y, 0.0f);
        float h2 = fmaxf(a.z + b.z, 0.0f);
        float h3 = fmaxf(a.w + b.w, 0.0f);
        float o0 = h0 * w01.x + h1 * w01.z + h2 * w23.x + h3 * w23.z;
        float o1 = h0 * w01.y + h1 * w01.w + h2 * w23.y + h3 * w23.w;
#pragma unroll
        for (int off = 16; off > 0; off >>= 1) {
            o0 += __shfl_xor(o0, off, 32);
            o1 += __shfl_xor(o1, off, 32);
        }
        if (lane == 0) {
            float v0 = o0 + c0;
            float v1 = o1 + c1;
            float m  = fmaxf(v0, v1);
            float e0 = __expf(v0 - m);
            float e1 = __expf(v1 - m);
            float s  = e0 + e1;
            size_t p = (size_t)t * G + g;
            out[p * 2 + 0] = e0 / s;
            out[p * 2 + 1] = e1 / s;
        }
    }
}

// ---------------------------------------------------------------------------
// Host orchestration
// ---------------------------------------------------------------------------
static inline size_t align_up(size_t x) { return (x + 255) & ~(size_t)255; }

extern "C" void kernel_launch(void* const* d_in, const int* in_sizes, int n_in,
                              void* d_out, int out_size, void* d_ws, size_t ws_size,
                              hipStream_t stream) {
    const float* x    = (const float*)d_in[0];
    const float* W1   = (const float*)d_in[1];
    const float* b1   = (const float*)d_in[2];
    const float* W2   = (const float*)d_in[3];
    const float* b2   = (const float*)d_in[4];
    const float* Wm1  = (const float*)d_in[5];
    const float* bm1  = (const float*)d_in[6];
    const float* Wm2  = (const float*)d_in[7];
    const float* bm2  = (const float*)d_in[8];
    const int*   ei   = (const int*)d_in[9];
    const int*   nump = (const int*)d_in[10];
    const int*   tfi  = (const int*)d_in[11];
    const int*   gi   = (const int*)d_in[12];

    const int Nn   = in_sizes[0] / 128;  // 80000 nodes
    const int E    = in_sizes[9] / 2;    // 2,560,000 edges
    const int T    = in_sizes[11];       // 128 TFs
    const int G    = in_sizes[12];       // 8000 genes
    const int DEMB = in_sizes[3] / 128;  // 64 (W2 is 128 x D_EMB)

    // Workspace layout
    char*  ws   = (char*)d_ws;
    float* deg  = (float*)ws;  ws += align_up((size_t)Nn * sizeof(float));
    float* bufA = (float*)ws;  ws += align_up((size_t)Nn * D_FEAT * sizeof(float)); // agg1 / agg2
    float* bufH = (float*)ws;  ws += align_up((size_t)Nn * D_FEAT * sizeof(float)); // h, then emb
    float* Atf  = (float*)ws;  ws += align_up((size_t)T * D_FEAT * sizeof(float));
    float* Bg   = (float*)ws;  ws += align_up((size_t)G * D_FEAT * sizeof(float));

    // 1) degrees and first aggregation: agg1 = segsum(x[src] -> dst) / deg
    hipMemsetAsync(deg, 0, (size_t)Nn * sizeof(float), stream);
    hipMemsetAsync(bufA, 0, (size_t)Nn * D_FEAT * sizeof(float), stream);
    degree_kernel<<<10000, 256, 0, stream>>>(ei, deg, E);
    scatter_kernel<<<20000, 256, 0, stream>>>(x, ei, bufA, E);
    normalize_kernel<<<10000, 256, 0, stream>>>(bufA, deg, Nn);

    // 2) h = relu(agg1 @ W1 + b1)   [Nn,128] @ [128,128]
    {
        int blocks = ((Nn >> 4) + 7) / 8;
        gemm_wmma_kernel<8, true, false><<<blocks, 256, 0, stream>>>(
            bufA, W1, b1, bufH, Nn, D_FEAT, nullptr, nullptr);
    }

    // 3) second aggregation: agg2 = segsum(h[src] -> dst) / deg
    hipMemsetAsync(bufA, 0, (size_t)Nn * D_FEAT * sizeof(float), stream);
    scatter_kernel<<<20000, 256, 0, stream>>>(bufH, ei, bufA, E);
    normalize_kernel<<<10000, 256, 0, stream>>>(bufA, deg, Nn);

    // 4) emb = agg2 @ W2 + b2   [Nn,128] @ [128,64]  (reuse bufH for emb)
    float* emb = bufH;
    {
        int blocks = ((Nn >> 4) + 7) / 8;
        gemm_wmma_kernel<4, false, false><<<blocks, 256, 0, stream>>>(
            bufA, W2, b2, emb, Nn, D_FEAT, nullptr, nullptr);
    }

    // 5) Pair MLP factored over the concat:
    //    Atf[t,:] = emb[num+tfi[t],:] @ Wm1[0:64,:]  + bm1   (bias folded once)
    //    Bg [g,:] = emb[num+gi[g],:]  @ Wm1[64:128,:]
    {
        int blocks = ((T >> 4) + 7) / 8;
        gemm_wmma_kernel<8, false, true><<<blocks, 256, 0, stream>>>(
            emb, Wm1, bm1, Atf, T, DEMB, tfi, nump);
    }
    {
        int blocks = ((G >> 4) + 7) / 8;
        gemm_wmma_kernel<8, false, true><<<blocks, 256, 0, stream>>>(
            emb, Wm1 + (size_t)DEMB * D_FEAT, nullptr, Bg, G, DEMB, gi, nump);
    }

    // 6) out[t*G+g,:] = softmax(relu(Atf[t]+Bg[g]) @ Wm2 + bm2)
    pair_kernel<<<T * GCHUNKS, 256, 0, stream>>>(Atf, Bg, Wm2, bm2, (float*)d_out, T, G);
}